// FullyAdjacent_31971736551538
// MI455X (gfx1250) — compile-verified
//
#include <hip/hip_runtime.h>

// Output: rewired edge index, int32, laid out as
//   out[0 .. half)        = rows: (i mod N^2)/N  == (i/N) mod N   (tiled G times)
//   out[half .. 2*half)   = cols: j mod N        (j = i - half)
// half = G*N*N = out_size/2.  Pure store-bandwidth kernel (536 MB at the
// reference sizes): use b128 non-temporal stores, wave32-friendly sizing.

typedef __attribute__((ext_vector_type(4))) int v4i;

// Fast path: N is a power of two, half and total divisible by 4.
// Each thread writes aligned int4 chunks in a grid-stride loop.
__global__ void rewire_pow2_kernel(int* __restrict__ out,
                                   long long half, long long total,
                                   int shift, int mask) {
    const long long nchunks = total >> 2;
    const long long stride  = (long long)gridDim.x * blockDim.x;
    for (long long t = (long long)blockIdx.x * blockDim.x + threadIdx.x;
         t < nchunks; t += stride) {
        const long long i0 = t << 2;
        v4i vec;
        if (i0 < half) {
            // rows half: value constant across the aligned 4-chunk
            // (chunk cannot straddle a multiple-of-N boundary since 4 | N)
            const int v = (int)((i0 >> shift) & mask);
            vec.x = v; vec.y = v; vec.z = v; vec.w = v;
        } else {
            // cols half: consecutive values, no wrap inside aligned chunk
            const long long j = i0 - half;
            const int c = (int)(j & mask);
            vec.x = c; vec.y = c + 1; vec.z = c + 2; vec.w = c + 3;
        }
        // Output (536 MB) >> L2 (192 MB) and never re-read: non-temporal
        // b128 store -> global_store_b128 ... th:TH_STORE_NT on gfx1250.
        __builtin_nontemporal_store(vec, (v4i*)(out + i0));
    }
}

// Generic correctness fallback (non-pow2 N or unaligned sizes): scalar
// grid-stride, still non-temporal.
__global__ void rewire_generic_kernel(int* __restrict__ out,
                                      long long half, long long total,
                                      int N) {
    const long long stride = (long long)gridDim.x * blockDim.x;
    for (long long i = (long long)blockIdx.x * blockDim.x + threadIdx.x;
         i < total; i += stride) {
        int v;
        if (i < half) {
            v = (int)((i / N) % N);
        } else {
            v = (int)((i - half) % N);
        }
        __builtin_nontemporal_store(v, out + i);
    }
}

extern "C" void kernel_launch(void* const* d_in, const int* in_sizes, int n_in,
                              void* d_out, int out_size, void* d_ws, size_t ws_size,
                              hipStream_t stream) {
    (void)d_in; (void)n_in; (void)d_ws; (void)ws_size;

    // setup_inputs order: x (N*F), edge_index (2*E), batch (N)
    const int N = (n_in >= 3) ? in_sizes[2] : 4096;

    int* out = (int*)d_out;
    const long long total = (long long)out_size;   // 2*G*N*N
    const long long half  = total >> 1;            // G*N*N

    const bool pow2  = (N >= 4) && ((N & (N - 1)) == 0);
    const bool align = ((total & 3LL) == 0) && ((half & 3LL) == 0);

    const int threads = 256;                        // 8 wave32 waves per block

    if (pow2 && align) {
        int shift = 0;
        while ((1 << shift) < N) ++shift;           // log2(N)
        const int mask = N - 1;

        const long long nchunks = total >> 2;       // int4 chunks
        long long want = (nchunks + threads - 1) / threads;
        int blocks = (want > 8192) ? 8192 : (int)((want > 0) ? want : 1);

        rewire_pow2_kernel<<<blocks, threads, 0, stream>>>(out, half, total,
                                                           shift, mask);
    } else {
        long long want = (total + threads - 1) / threads;
        int blocks = (want > 8192) ? 8192 : (int)((want > 0) ? want : 1);

        rewire_generic_kernel<<<blocks, threads, 0, stream>>>(out, half, total, N);
    }
}